// PDEFunc_37589553774665
// MI455X (gfx1250) — compile-verified
//
#include <hip/hip_runtime.h>
#include <math.h>

// Problem constants (match reference)
#define KK   128
#define NXY  65536          // 256*256
#define MB   8
#define NT   32
#define DD   384            // 3*K
#define HH   128

typedef float v2f __attribute__((ext_vector_type(2)));
typedef float v8f __attribute__((ext_vector_type(8)));

// ---------------------------------------------------------------------------
// Kernel A: sequential RK4 (3/8 rule) neural ODE, single workgroup.
// Exploits batch-invariance: integrates ONE 384-dim state.
// Writes coeff[t][d] (t=0..31) into ws, plus transposed weights for coalesced
// L2 reads during the integration loop.
// ws layout (floats): [0)        coeff      NT*DD   = 12288
//                     [12288)    W1T (i,o)  DD*HH   = 49152
//                     [61440)    W2T (i,o)  HH*HH   = 16384
//                     [77824)    W3T (i,o)  HH*DD   = 49152
// ---------------------------------------------------------------------------
__device__ __forceinline__ void mlp_block(const float* __restrict__ cin,
                                          float* __restrict__ kout,
                                          const float* __restrict__ W1T,
                                          const float* __restrict__ b1,
                                          const float* __restrict__ W2T,
                                          const float* __restrict__ b2,
                                          const float* __restrict__ W3T,
                                          const float* __restrict__ b3,
                                          float* __restrict__ h1,
                                          float* __restrict__ h2,
                                          float* __restrict__ part) {
  const int tid = threadIdx.x;
  // Layer 1: 128 outputs, 4-way split over 384 inputs (96 each)
  {
    const int o = tid & 127, p = tid >> 7;     // p in 0..3
    const int i0 = p * 96;
    float acc = 0.f;
#pragma unroll 4
    for (int i = i0; i < i0 + 96; ++i) acc = fmaf(cin[i], W1T[i * HH + o], acc);
    part[tid] = acc;
  }
  __syncthreads();
  if (tid < HH) {
    float v = b1[tid] + part[tid] + part[tid + 128] + part[tid + 256] + part[tid + 384];
    h1[tid] = v > 0.f ? v : 0.f;               // ReLU
  }
  __syncthreads();
  // Layer 2: 128 outputs, 4-way split over 128 inputs (32 each)
  {
    const int o = tid & 127, p = tid >> 7;
    const int i0 = p * 32;
    float acc = 0.f;
#pragma unroll 4
    for (int i = i0; i < i0 + 32; ++i) acc = fmaf(h1[i], W2T[i * HH + o], acc);
    part[tid] = acc;
  }
  __syncthreads();
  if (tid < HH) {
    float v = b2[tid] + part[tid] + part[tid + 128] + part[tid + 256] + part[tid + 384];
    h2[tid] = v > 0.f ? v : expm1f(v);         // ELU (alpha=1)
  }
  __syncthreads();
  // Layer 3: 384 outputs, one thread each, 128-long dot
  if (tid < DD) {
    float acc = b3[tid];
#pragma unroll 4
    for (int i = 0; i < HH; ++i) acc = fmaf(h2[i], W3T[i * DD + tid], acc);
    kout[tid] = acc;
  }
  __syncthreads();
}

__global__ __launch_bounds__(512) void ode_rk4_kernel(
    const float* __restrict__ t, const float* __restrict__ init,
    const float* __restrict__ W1, const float* __restrict__ b1,
    const float* __restrict__ W2, const float* __restrict__ b2,
    const float* __restrict__ W3, const float* __restrict__ b3,
    float* __restrict__ ws) {
  float* coeff = ws;
  float* W1T = ws + 12288;
  float* W2T = W1T + DD * HH;
  float* W3T = W2T + HH * HH;
  const int tid = threadIdx.x;

  // One-time weight transposes (coalesced-read phase afterwards)
  for (int e = tid; e < HH * DD; e += 512) {  // W1 (HH rows x DD cols)
    int o = e / DD, i = e % DD;
    W1T[i * HH + o] = W1[e];
  }
  for (int e = tid; e < HH * HH; e += 512) {  // W2
    int o = e / HH, i = e % HH;
    W2T[i * HH + o] = W2[e];
  }
  for (int e = tid; e < DD * HH; e += 512) {  // W3 (DD rows x HH cols)
    int o = e / HH, i = e % HH;
    W3T[i * DD + o] = W3[e];
  }

  __shared__ float y[DD], cin[DD], k1[DD], k2[DD], k3[DD], k4[DD];
  __shared__ float h1[HH], h2[HH], part[512];

  if (tid < DD) {
    float v = init[tid];
    y[tid] = v;
    coeff[tid] = v;                            // coeff[t=0] = y0
  }
  __syncthreads();                             // also fences W*T global writes (block scope)

  for (int s = 0; s < NT - 1; ++s) {
    const float dt = t[s + 1] - t[s];

    // k1 = f(y)
    mlp_block(y, k1, W1T, b1, W2T, b2, W3T, b3, h1, h2, part);
    // k2 = f(y + dt*k1/3)
    if (tid < DD) cin[tid] = fmaf(dt * (1.f / 3.f), k1[tid], y[tid]);
    __syncthreads();
    mlp_block(cin, k2, W1T, b1, W2T, b2, W3T, b3, h1, h2, part);
    // k3 = f(y + dt*(k2 - k1/3))
    if (tid < DD) cin[tid] = y[tid] + dt * (k2[tid] - (1.f / 3.f) * k1[tid]);
    __syncthreads();
    mlp_block(cin, k3, W1T, b1, W2T, b2, W3T, b3, h1, h2, part);
    // k4 = f(y + dt*(k1 - k2 + k3))
    if (tid < DD) cin[tid] = y[tid] + dt * (k1[tid] - k2[tid] + k3[tid]);
    __syncthreads();
    mlp_block(cin, k4, W1T, b1, W2T, b2, W3T, b3, h1, h2, part);

    if (tid < DD) {
      float v = y[tid] + dt * 0.125f * (k1[tid] + 3.f * (k2[tid] + k3[tid]) + k4[tid]);
      y[tid] = v;
      coeff[(s + 1) * DD + tid] = v;
    }
    __syncthreads();
  }
}

// ---------------------------------------------------------------------------
// Kernel B: out[t,m,c,p] = sum_k coeff[t, 3k+c] * basis[k,c,p], m broadcast.
// Per channel c: GEMM A(32x128) @ B(128x65536) in fp32 via
// V_WMMA_F32_16X16X4_F32. Each wave owns 16 pixels and both 16-row M tiles.
// A tile lives in LDS (rows padded to 132 words -> conflict-free ds_load_b64);
// B fragments stream directly from global (basis read exactly once from HBM,
// so loads use TH=NT), and out is written exactly once (stores use TH=NT) to
// keep the 300 MB of one-shot traffic from churning L2.
// ---------------------------------------------------------------------------
__global__ __launch_bounds__(256) void einsum_wmma_kernel(
    const float* __restrict__ coeff,  // ws: NT x DD
    const float* __restrict__ basis,  // K x 3 x 65536
    float* __restrict__ out) {        // NT x MB x 3 x 65536
  const int c  = blockIdx.x;          // channel 0..2
  const int pb = blockIdx.y;          // pixel block 0..511 (128 pixels each)
  __shared__ float sA[NT * 132];

  const int tid = threadIdx.x;
  // Stage A tile: sA[t][k] = coeff[t*384 + 3k + c]
  for (int e = tid; e < NT * KK; e += 256) {
    const int tt = e >> 7, k = e & 127;
    sA[tt * 132 + k] = coeff[tt * DD + 3 * k + c];
  }
  __syncthreads();

  const int lane  = tid & 31;
  const int wave  = tid >> 5;
  const int nlane = lane & 15;
  const int khi   = (lane >> 4) << 1;       // lanes 16-31 hold K+2,K+3
  const int pix0  = pb * 128 + wave * 16;
  const size_t krow = (size_t)3 * NXY;      // element stride between k-rows
  const float* bbase = basis + (size_t)c * NXY + pix0 + nlane;

  v8f acc0 = {};  // M rows 0..15  (t = 0..15)
  v8f acc1 = {};  // M rows 16..31 (t = 16..31)

#pragma unroll 4
  for (int k0 = 0; k0 < KK; k0 += 4) {
    const v2f a0 = *(const v2f*)&sA[nlane * 132 + k0 + khi];
    const v2f a1 = *(const v2f*)&sA[(16 + nlane) * 132 + k0 + khi];
    const float* bp = bbase + (size_t)(k0 + khi) * krow;
    v2f b;
    b.x = __builtin_nontemporal_load(bp);          // TH=NT: streamed once
    b.y = __builtin_nontemporal_load(bp + krow);
    acc0 = __builtin_amdgcn_wmma_f32_16x16x4_f32(false, a0, false, b,
                                                 (short)0, acc0, false, false);
    acc1 = __builtin_amdgcn_wmma_f32_16x16x4_f32(false, a1, false, b,
                                                 (short)0, acc1, false, false);
  }

  // C layout: VGPR r -> (M = r + 8*(lane>=16), N = lane&15). Broadcast over m.
  const int thalf = (lane >> 4) * 8;
  const size_t pix = (size_t)pix0 + nlane;
#pragma unroll
  for (int r = 0; r < 8; ++r) {
    const int t0 = r + thalf;        // from acc0
    const int t1 = t0 + 16;          // from acc1
#pragma unroll
    for (int m = 0; m < MB; ++m) {
      __builtin_nontemporal_store(
          acc0[r], &out[(((size_t)(t0 * MB + m)) * 3 + c) * NXY + pix]);
      __builtin_nontemporal_store(
          acc1[r], &out[(((size_t)(t1 * MB + m)) * 3 + c) * NXY + pix]);
    }
  }
}

// ---------------------------------------------------------------------------
// Host-side launcher. Inputs (setup_inputs order):
// 0 grid0 (unused), 1 t, 2 init_coeffs, 3 W1, 4 b1, 5 W2, 6 b2, 7 W3, 8 b3,
// 9 basis. Output: (32, 8, 3, 256, 256) float32.
// ---------------------------------------------------------------------------
extern "C" void kernel_launch(void* const* d_in, const int* in_sizes, int n_in,
                              void* d_out, int out_size, void* d_ws, size_t ws_size,
                              hipStream_t stream) {
  const float* t    = (const float*)d_in[1];
  const float* init = (const float*)d_in[2];
  const float* W1   = (const float*)d_in[3];
  const float* b1   = (const float*)d_in[4];
  const float* W2   = (const float*)d_in[5];
  const float* b2   = (const float*)d_in[6];
  const float* W3   = (const float*)d_in[7];
  const float* b3   = (const float*)d_in[8];
  const float* basis = (const float*)d_in[9];
  float* ws  = (float*)d_ws;
  float* out = (float*)d_out;

  ode_rk4_kernel<<<1, 512, 0, stream>>>(t, init, W1, b1, W2, b2, W3, b3, ws);
  einsum_wmma_kernel<<<dim3(3, 512), 256, 0, stream>>>(ws, basis, out);
}